// GCL_GCN_23055384445693
// MI455X (gfx1250) — compile-verified
//
#include <hip/hip_runtime.h>
#include <hip/hip_bf16.h>
#include <math.h>

#define N_NODES 131072
#define HID 512
#define N_EDGES 131072

typedef __attribute__((ext_vector_type(2))) float v2f;
typedef __attribute__((ext_vector_type(8))) float v8f;

__device__ __forceinline__ float elu1(float x) {
    return x > 0.0f ? x : (expf(x) - 1.0f);
}

// ---------------------------------------------------------------------------
// WMMA fp32 GEMM:  C[M,N] = act(A[M,K] @ W[K,N] + bias[N])
// block = 128 threads (4 waves). Each wave -> one 16(M) x 64(N) strip.
// grid = (M/16, N/256). Uses V_WMMA_F32_16X16X4_F32 (full fp32 matrix core).
// ---------------------------------------------------------------------------
__global__ __launch_bounds__(128) void wmma_gemm_bias_act(
    const float* __restrict__ A, const float* __restrict__ W,
    const float* __restrict__ bias, float* __restrict__ C,
    int K, int N, int act)
{
    const int lane = threadIdx.x & 31;
    const int wave = threadIdx.x >> 5;
    const int row  = lane & 15;      // A row / B,C column within tile
    const int half = lane >> 4;      // lane half selects K pair / M half

    const int mbase = blockIdx.x * 16;
    const int nbase = blockIdx.y * 256 + wave * 64;

    v8f acc0 = {}, acc1 = {}, acc2 = {}, acc3 = {};

    const float* arow = A + (size_t)(mbase + row) * K + 2 * half;
    const int kq = 2 * half;

    for (int k = 0; k < K; k += 4) {
        // A fragment: lanes 0-15 hold K=k,k+1; lanes 16-31 hold K=k+2,k+3
        v2f a = *(const v2f*)(arow + k);

        const float* wp = W + (size_t)(k + kq) * N + nbase + row;
        v2f b0; b0.x = wp[0];  b0.y = wp[N];
        v2f b1; b1.x = wp[16]; b1.y = wp[N + 16];
        v2f b2; b2.x = wp[32]; b2.y = wp[N + 32];
        v2f b3; b3.x = wp[48]; b3.y = wp[N + 48];

        acc0 = __builtin_amdgcn_wmma_f32_16x16x4_f32(false, a, false, b0, (short)0, acc0, false, false);
        acc1 = __builtin_amdgcn_wmma_f32_16x16x4_f32(false, a, false, b1, (short)0, acc1, false, false);
        acc2 = __builtin_amdgcn_wmma_f32_16x16x4_f32(false, a, false, b2, (short)0, acc2, false, false);
        acc3 = __builtin_amdgcn_wmma_f32_16x16x4_f32(false, a, false, b3, (short)0, acc3, false, false);
    }

    v8f accs[4] = {acc0, acc1, acc2, acc3};
    #pragma unroll
    for (int nt = 0; nt < 4; ++nt) {
        const int ocol = nbase + nt * 16 + row;
        const float bv = bias[ocol];
        #pragma unroll
        for (int r = 0; r < 8; ++r) {
            const int orow = mbase + r + 8 * half;  // VGPR r -> M=r (lo half), M=r+8 (hi half)
            float v = accs[nt][r] + bv;
            if (act) v = elu1(v);
            C[(size_t)orow * N + ocol] = v;
        }
    }
}

// ---------------------------------------------------------------------------
// Graph helpers
// ---------------------------------------------------------------------------
__global__ void init_deg_k(float* __restrict__ dout, float* __restrict__ din) {
    int i = blockIdx.x * blockDim.x + threadIdx.x;
    if (i < N_NODES) { dout[i] = 1.0f; din[i] = 1.0f; }  // self-loop counts
}

__global__ void deg_count_k(const int* __restrict__ src, const int* __restrict__ dst,
                            float* __restrict__ dout, float* __restrict__ din) {
    int e = blockIdx.x * blockDim.x + threadIdx.x;
    if (e < N_EDGES) {
        atomicAdd(&dout[src[e]], 1.0f);
        atomicAdd(&din[dst[e]], 1.0f);
    }
}

__global__ void rsqrt_k(float* __restrict__ p, int n) {
    int i = blockIdx.x * blockDim.x + threadIdx.x;
    if (i < n) p[i] = rsqrtf(p[i]);   // deg >= 1 always (self-loops) => clip is a no-op
}

__global__ void zero_k(float4* __restrict__ p, int n4) {
    int i = blockIdx.x * blockDim.x + threadIdx.x;
    if (i < n4) p[i] = make_float4(0.f, 0.f, 0.f, 0.f);
}

// Scatter-add: agg[dst] += h[src] * norm_out[src].  2 edges per 256-thread block,
// 128 threads x float4 cover the 512-wide feature row.
__global__ __launch_bounds__(256) void scatter_k(
    const float* __restrict__ h, const float* __restrict__ nout,
    const int* __restrict__ src, const int* __restrict__ dst,
    float* __restrict__ agg)
{
    const int e = blockIdx.x * 2 + (threadIdx.x >> 7);
    const int f = (threadIdx.x & 127) << 2;
    const int s = src[e];
    const int d = dst[e];
    const float ns = nout[s];
    const float4 v = *(const float4*)(h + (size_t)s * HID + f);
    float* ap = agg + (size_t)d * HID + f;
    atomicAdd(ap + 0, v.x * ns);
    atomicAdd(ap + 1, v.y * ns);
    atomicAdd(ap + 2, v.z * ns);
    atomicAdd(ap + 3, v.w * ns);
}

// Layer 0 finalize (in place into h): h = elu((agg + h*norm_out)*norm_in + b)
__global__ __launch_bounds__(256) void finalize0_k(
    float* __restrict__ h, const float* __restrict__ agg,
    const float* __restrict__ nout, const float* __restrict__ nin,
    const float* __restrict__ bias)
{
    const int idx  = blockIdx.x * blockDim.x + threadIdx.x;  // one float4 each
    const int node = idx >> 7;
    const int f    = (idx & 127) << 2;
    const float no = nout[node], ni = nin[node];
    const size_t off = (size_t)node * HID + f;
    const float4 hv = *(const float4*)(h + off);
    const float4 av = *(const float4*)(agg + off);
    const float4 bv = *(const float4*)(bias + f);
    float4 o;
    o.x = elu1((av.x + hv.x * no) * ni + bv.x);
    o.y = elu1((av.y + hv.y * no) * ni + bv.y);
    o.z = elu1((av.z + hv.z * no) * ni + bv.z);
    o.w = elu1((av.w + hv.w * no) * ni + bv.w);
    *(float4*)(h + off) = o;
}

// Layer 1 pre-GEMM (in place into agg): agg = (agg + h*norm_out)*norm_in
__global__ __launch_bounds__(256) void pre1_k(
    const float* __restrict__ h, float* __restrict__ agg,
    const float* __restrict__ nout, const float* __restrict__ nin)
{
    const int idx  = blockIdx.x * blockDim.x + threadIdx.x;
    const int node = idx >> 7;
    const int f    = (idx & 127) << 2;
    const float no = nout[node], ni = nin[node];
    const size_t off = (size_t)node * HID + f;
    const float4 hv = *(const float4*)(h + off);
    float4 av = *(const float4*)(agg + off);
    av.x = (av.x + hv.x * no) * ni;
    av.y = (av.y + hv.y * no) * ni;
    av.z = (av.z + hv.z * no) * ni;
    av.w = (av.w + hv.w * no) * ni;
    *(float4*)(agg + off) = av;
}

// ---------------------------------------------------------------------------
extern "C" void kernel_launch(void* const* d_in, const int* in_sizes, int n_in,
                              void* d_out, int out_size, void* d_ws, size_t ws_size,
                              hipStream_t stream) {
    (void)in_sizes; (void)n_in; (void)out_size; (void)ws_size;

    // setup_inputs() dict order
    const float* feat0 = (const float*)d_in[0];
    const float* fc0_w = (const float*)d_in[1];
    const float* fc0_b = (const float*)d_in[2];
    const float* feat1 = (const float*)d_in[3];
    const float* fc1_w = (const float*)d_in[4];
    const float* fc1_b = (const float*)d_in[5];
    const float* feat2 = (const float*)d_in[6];
    const float* fc2_w = (const float*)d_in[7];
    const float* fc2_b = (const float*)d_in[8];
    const float* gc0_b = (const float*)d_in[9];
    const float* gc1_w = (const float*)d_in[10];
    const float* gc1_b = (const float*)d_in[11];
    const int*   src   = (const int*)d_in[12];
    const int*   dst   = (const int*)d_in[13];
    float* out = (float*)d_out;

    // workspace: norms (2*N), h (N*HID), agg (N*HID)  ~ 538 MB
    float* nout = (float*)d_ws;
    float* nin  = nout + N_NODES;
    float* h    = nin + N_NODES;
    float* agg  = h + (size_t)N_NODES * HID;

    const int hid4 = N_NODES * (HID / 4);   // float4 count of a [N,HID] buffer

    // 1) degree norms (self-loops folded in as the init value 1.0)
    init_deg_k<<<N_NODES / 256, 256, 0, stream>>>(nout, nin);
    deg_count_k<<<N_EDGES / 256, 256, 0, stream>>>(src, dst, nout, nin);
    rsqrt_k<<<(2 * N_NODES) / 256, 256, 0, stream>>>(nout, 2 * N_NODES);

    // 2) per-type projections -> h  (fp32 WMMA, bias, no activation)
    dim3 blk(128);
    wmma_gemm_bias_act<<<dim3(65536 / 16, 2), blk, 0, stream>>>(feat0, fc0_w, fc0_b, h, 256, HID, 0);
    wmma_gemm_bias_act<<<dim3(32768 / 16, 2), blk, 0, stream>>>(feat1, fc1_w, fc1_b,
        h + (size_t)65536 * HID, 512, HID, 0);
    wmma_gemm_bias_act<<<dim3(32768 / 16, 2), blk, 0, stream>>>(feat2, fc2_w, fc2_b,
        h + (size_t)98304 * HID, 1024, HID, 0);

    // 3) layer 0: scatter-add then finalize (+self-loop term, norms, bias, ELU)
    zero_k<<<hid4 / 256, 256, 0, stream>>>((float4*)agg, hid4);
    scatter_k<<<N_EDGES / 2, 256, 0, stream>>>(h, nout, src, dst, agg);
    finalize0_k<<<(N_NODES * (HID / 4)) / 256, 256, 0, stream>>>(h, agg, nout, nin, gc0_b);

    // 4) layer 1: scatter-add, normalize, then W-GEMM + bias + ELU -> d_out
    zero_k<<<hid4 / 256, 256, 0, stream>>>((float4*)agg, hid4);
    scatter_k<<<N_EDGES / 2, 256, 0, stream>>>(h, nout, src, dst, agg);
    pre1_k<<<(N_NODES * (HID / 4)) / 256, 256, 0, stream>>>(h, agg, nout, nin);
    wmma_gemm_bias_act<<<dim3(N_NODES / 16, 2), blk, 0, stream>>>(agg, gc1_w, gc1_b, out, HID, HID, 1);
}